// ImageSSIMLoss_45878840656468
// MI455X (gfx1250) — compile-verified
//
#include <hip/hip_runtime.h>

typedef float v8f __attribute__((ext_vector_type(8)));
typedef float v2f __attribute__((ext_vector_type(2)));

// Only meaningful to check amdgcn builtins during the device pass.
#if defined(__HIP_DEVICE_COMPILE__)
#if !__has_builtin(__builtin_amdgcn_wmma_f32_16x16x4_f32)
#error "missing __builtin_amdgcn_wmma_f32_16x16x4_f32 on this toolchain"
#endif
#endif

#define IMGS        96          // 32 * 3
#define HW          512
#define PIX         (HW * HW)   // 262144
#define SEGS        32          // min/max segments per image
#define SEGLEN      (PIX / SEGS) // 8192
#define RS_PER_IMG  32          // 512 / 16 row strips
#define CS_PER_IMG  4           // 512 / 128 col strips
#define BLOCKS2     (IMGS * RS_PER_IMG * CS_PER_IMG) // 12288
#define EPS_SSIM    1e-6f
#define INV121      (1.0f / 121.0f)
#define HALO        5
#define RROWS       26          // 16 + 2*5 halo rows in region
#define RCOLS       140         // 138 used (128 + 2*5), padded stride
#define NCH         7           // K chunks of 4 (K padded 26 -> 28)
#define HROWS       28
#define HSTR        17          // padded col stride of H staging
#define NPIX_TOTAL  25165824.0f // 32*3*512*512

// ---------------- Kernel 1: per-segment min/max (float4 streaming) ----------
__global__ void ssim_range_partial(const float4* __restrict__ p4,
                                   const float4* __restrict__ t4,
                                   float* __restrict__ pmax,
                                   float* __restrict__ pmin) {
    int blk = blockIdx.x;                       // 0 .. IMGS*SEGS-1
    size_t base = (size_t)blk * (SEGLEN / 4);
    float mx = -3.4e38f, mn = 3.4e38f;
    for (int i = threadIdx.x; i < SEGLEN / 4; i += blockDim.x) {
        float4 a = p4[base + i];
        float4 b = t4[base + i];
        mx = fmaxf(mx, fmaxf(fmaxf(a.x, a.y), fmaxf(a.z, a.w)));
        mx = fmaxf(mx, fmaxf(fmaxf(b.x, b.y), fmaxf(b.z, b.w)));
        mn = fminf(mn, fminf(fminf(a.x, a.y), fminf(a.z, a.w)));
        mn = fminf(mn, fminf(fminf(b.x, b.y), fminf(b.z, b.w)));
    }
    for (int m = 16; m >= 1; m >>= 1) {
        mx = fmaxf(mx, __shfl_xor(mx, m, 32));
        mn = fminf(mn, __shfl_xor(mn, m, 32));
    }
    __shared__ float smx[8], smn[8];
    int wave = threadIdx.x >> 5;
    if ((threadIdx.x & 31) == 0) { smx[wave] = mx; smn[wave] = mn; }
    __syncthreads();
    if (threadIdx.x == 0) {
        float MX = smx[0], MN = smn[0];
        for (int i = 1; i < 8; ++i) { MX = fmaxf(MX, smx[i]); MN = fminf(MN, smn[i]); }
        pmax[blk] = MX; pmin[blk] = MN;
    }
}

// ---------------- Kernel 2: fold segments -> per-image c1, c2 ---------------
__global__ void ssim_range_final(const float* __restrict__ pmax,
                                 const float* __restrict__ pmin,
                                 float* __restrict__ c1buf,
                                 float* __restrict__ c2buf) {
    int img = blockIdx.x;   // one wave32 per image
    float mx = pmax[img * SEGS + threadIdx.x];
    float mn = pmin[img * SEGS + threadIdx.x];
    for (int m = 16; m >= 1; m >>= 1) {
        mx = fmaxf(mx, __shfl_xor(mx, m, 32));
        mn = fminf(mn, __shfl_xor(mn, m, 32));
    }
    if (threadIdx.x == 0) {
        float dr = fmaxf(mx - mn, EPS_SSIM);
        float c1 = 0.01f * dr; c1 *= c1;
        float c2 = 0.03f * dr; c2 *= c2;
        c1buf[img] = c1; c2buf[img] = c2;
    }
}

// ---------------- Kernel 3: fused WMMA box-filter SSIM ----------------------
// Block = 8 waves; each wave owns a 16x16 output tile; block covers 16x128.
// Separable 11x11 box sum as banded GEMM on the f32 WMMA pipe.
__global__ __launch_bounds__(256)
void ssim_tile_kernel(const float* __restrict__ pred,
                      const float* __restrict__ targ,
                      const float* __restrict__ c1buf,
                      const float* __restrict__ c2buf,
                      float* __restrict__ partial) {
    __shared__ float sP[RROWS][RCOLS];
    __shared__ float sT[RROWS][RCOLS];
    __shared__ float sH[8][HROWS][HSTR];   // per-wave D->B relayout staging
    __shared__ float swsum[8];

    int b   = blockIdx.x;
    int img = b / (RS_PER_IMG * CS_PER_IMG);
    int rem = b % (RS_PER_IMG * CS_PER_IMG);
    int R   = (rem / CS_PER_IMG) * 16;     // output row base
    int C0  = (rem % CS_PER_IMG) * 128;    // output col base of the strip

    const float* p = pred + (size_t)img * PIX;
    const float* t = targ + (size_t)img * PIX;

    // Cooperative halo load (zero padding == SAME conv with zero pad)
    for (int idx = threadIdx.x; idx < RROWS * RCOLS; idx += 256) {
        int r = idx / RCOLS, c = idx % RCOLS;
        int gr = R - HALO + r;
        int gc = C0 - HALO + c;
        float pv = 0.0f, tv = 0.0f;
        if (gr >= 0 && gr < HW && gc >= 0 && gc < HW) {
            int o = gr * HW + gc;
            pv = p[o]; tv = t[o];
        }
        sP[r][c] = pv; sT[r][c] = tv;
    }
    __syncthreads();

    int wave = threadIdx.x >> 5;
    int lane = threadIdx.x & 31;
    int m    = lane & 15;     // M (A) or N (B/D) index for this lane
    int lh   = lane >> 4;     // lane half: selects K pair within a chunk
    int lc0  = wave * 16;     // LDS col of global col (C-5) for this wave

    // Hoist raw A-side data: rows {tt*10+m}, cols lc0 + 4j + 2*lh (+1)
    float pa[2][NCH][2], ta[2][NCH][2];
#pragma unroll
    for (int tt = 0; tt < 2; ++tt) {
        int rr = tt * 10 + m;
#pragma unroll
        for (int j = 0; j < NCH; ++j) {
            int cc = lc0 + 4 * j + 2 * lh;
            pa[tt][j][0] = sP[rr][cc]; pa[tt][j][1] = sP[rr][cc + 1];
            ta[tt][j][0] = sT[rr][cc]; ta[tt][j][1] = sT[rr][cc + 1];
        }
    }

    // Band weights: 1.0 iff (k - n) in [0,10]; same per-lane formula serves as
    // horizontal-pass B operand and vertical-pass A operand.
    v2f wband[NCH];
#pragma unroll
    for (int j = 0; j < NCH; ++j) {
        int kk = 4 * j + 2 * lh;
        wband[j].x = (kk - m >= 0 && kk - m <= 10) ? 1.0f : 0.0f;
        wband[j].y = (kk + 1 - m >= 0 && kk + 1 - m <= 10) ? 1.0f : 0.0f;
    }

    v8f res[5];
#pragma unroll
    for (int q = 0; q < 5; ++q) {
        // zero K-pad rows 26,27 of staging
        sH[wave][26 + lh][m] = 0.0f;

        // Horizontal pass: H(rows of region) = X(16x28) x Wband(28x16)
#pragma unroll
        for (int tt = 0; tt < 2; ++tt) {
            v8f acc = {};
#pragma unroll
            for (int j = 0; j < NCH; ++j) {
                float p0 = pa[tt][j][0], p1 = pa[tt][j][1];
                float t0 = ta[tt][j][0], t1 = ta[tt][j][1];
                v2f a;
                if      (q == 0) { a.x = p0;      a.y = p1;      }
                else if (q == 1) { a.x = t0;      a.y = t1;      }
                else if (q == 2) { a.x = p0 * p0; a.y = p1 * p1; }
                else if (q == 3) { a.x = t0 * t0; a.y = t1 * t1; }
                else             { a.x = p0 * t0; a.y = p1 * t1; }
#if defined(__HIP_DEVICE_COMPILE__)
                acc = __builtin_amdgcn_wmma_f32_16x16x4_f32(
                    false, a, false, wband[j], (short)0, acc, false, false);
#endif
            }
            // D layout: VGPR i -> region row tt*10 + i + 8*lh, col m
#pragma unroll
            for (int i = 0; i < 8; ++i)
                sH[wave][tt * 10 + i + 8 * lh][m] = acc[i];
        }
        __syncthreads();   // uniform across block; publishes H relayout

        // Vertical pass: V(16x16) = Wband(16x28) x H(28x16)
        v8f vacc = {};
#pragma unroll
        for (int j = 0; j < NCH; ++j) {
            int kk = 4 * j + 2 * lh;
            v2f bb;
            bb.x = sH[wave][kk][m];
            bb.y = sH[wave][kk + 1][m];
#if defined(__HIP_DEVICE_COMPILE__)
            vacc = __builtin_amdgcn_wmma_f32_16x16x4_f32(
                false, wband[j], false, bb, (short)0, vacc, false, false);
#endif
        }
#pragma unroll
        for (int i = 0; i < 8; ++i) res[q][i] = vacc[i] * INV121;
        __syncthreads();   // protect sH reuse for next quantity
    }

    // Per-pixel SSIM math on the 8 pixels this lane owns (D layout)
    float c1 = c1buf[img], c2 = c2buf[img];
    float lsum = 0.0f;
#pragma unroll
    for (int i = 0; i < 8; ++i) {
        float mp  = res[0][i], mt  = res[1][i];
        float epp = res[2][i], ett = res[3][i], ept = res[4][i];
        float sp  = fmaxf(epp - mp * mp, 0.0f);
        float st  = fmaxf(ett - mt * mt, 0.0f);
        float sx  = ept - mp * mt;
        float num = (2.0f * mp * mt + c1) * (2.0f * sx + c2);
        float den = (mp * mp + mt * mt + c1) * (sp + st + c2);
        float ssim = num / (den + EPS_SSIM);
        lsum += fmaxf((1.0f - ssim) * 0.5f, 0.0f);
    }
    for (int mk = 16; mk >= 1; mk >>= 1) lsum += __shfl_xor(lsum, mk, 32);
    if (lane == 0) swsum[wave] = lsum;
    __syncthreads();
    if (threadIdx.x == 0) {
        float s = 0.0f;
        for (int i = 0; i < 8; ++i) s += swsum[i];
        partial[blockIdx.x] = s;
    }
}

// ---------------- Kernel 4: deterministic final reduction -------------------
__global__ void ssim_finalize(const float* __restrict__ partial,
                              float* __restrict__ out, int n) {
    float s = 0.0f;
    for (int i = threadIdx.x; i < n; i += 256) s += partial[i];
    for (int mk = 16; mk >= 1; mk >>= 1) s += __shfl_xor(s, mk, 32);
    __shared__ float sw[8];
    if ((threadIdx.x & 31) == 0) sw[threadIdx.x >> 5] = s;
    __syncthreads();
    if (threadIdx.x == 0) {
        float tot = 0.0f;
        for (int i = 0; i < 8; ++i) tot += sw[i];
        out[0] = tot / NPIX_TOTAL;
    }
}

extern "C" void kernel_launch(void* const* d_in, const int* in_sizes, int n_in,
                              void* d_out, int out_size, void* d_ws, size_t ws_size,
                              hipStream_t stream) {
    (void)in_sizes; (void)n_in; (void)out_size; (void)ws_size;
    const float* pred = (const float*)d_in[0];
    const float* targ = (const float*)d_in[1];
    float* out = (float*)d_out;
    float* ws  = (float*)d_ws;

    float* c1buf   = ws;                    // 96
    float* c2buf   = ws + IMGS;             // 96
    float* pmax    = ws + 2 * IMGS;         // 3072
    float* pmin    = pmax + IMGS * SEGS;    // 3072
    float* partial = pmin + IMGS * SEGS;    // 12288

    ssim_range_partial<<<IMGS * SEGS, 256, 0, stream>>>(
        (const float4*)pred, (const float4*)targ, pmax, pmin);
    ssim_range_final<<<IMGS, 32, 0, stream>>>(pmax, pmin, c1buf, c2buf);
    ssim_tile_kernel<<<BLOCKS2, 256, 0, stream>>>(pred, targ, c1buf, c2buf, partial);
    ssim_finalize<<<1, 256, 0, stream>>>(partial, out, BLOCKS2);
}